// Dynamic_conv2d_first_39694087749951
// MI455X (gfx1250) — compile-verified
//
#include <hip/hip_runtime.h>
#include <hip/hip_bf16.h>

typedef __attribute__((ext_vector_type(2))) float v2f;
typedef __attribute__((ext_vector_type(8))) float v8f;

#define HW_H 224
#define HW_W 224
#define CIN 3
#define OUTC 64
#define KATT 10
#define BATCH 16
#define KK27 27
#define TEMP_INV (1.0f / 34.0f)

// ---------------------------------------------------------------------------
// Kernel 1: global average pool. One block per (b, c) pair -> pooled[48].
// ---------------------------------------------------------------------------
__global__ __launch_bounds__(256)
void pool_kernel(const float* __restrict__ x, float* __restrict__ pooled) {
    __shared__ float sd[256];
    const int bc = blockIdx.x;                 // 0..47  (b*3 + c)
    const float* p = x + (size_t)bc * (HW_H * HW_W);
    float s = 0.0f;
    for (int i = threadIdx.x; i < HW_H * HW_W; i += 256) s += p[i];
    sd[threadIdx.x] = s;
    __syncthreads();
    for (int off = 128; off > 0; off >>= 1) {
        if (threadIdx.x < off) sd[threadIdx.x] += sd[threadIdx.x + off];
        __syncthreads();
    }
    if (threadIdx.x == 0) pooled[bc] = sd[0] * (1.0f / (float)(HW_H * HW_W));
}

// ---------------------------------------------------------------------------
// Kernel 2: attention (softmax over K=10) + WMMA aggregation of kernels/bias.
// One wave (32 threads) per 16-wide column tile.
//   blocks 0..107   : agg_w[b][n] = sum_k att[b][k] * weight_3c[k][n], n<1728
//   blocks 108..111 : agg_b[b][n] = sum_k att[b][k] * bias_k[k][n],    n<64
// GEMM: M=16 batches, K=10 (padded to 12), N=16 per block, fp32 WMMA 16x16x4.
// ---------------------------------------------------------------------------
__global__ __launch_bounds__(32)
void attn_agg_kernel(const float* __restrict__ pooled,
                     const float* __restrict__ w1,   // [10][3]
                     const float* __restrict__ w2,   // [10][10]
                     const float* __restrict__ b2,   // [10]
                     const float* __restrict__ wc,   // [10][1728]
                     const float* __restrict__ bk,   // [10][64]
                     float* __restrict__ agg_w,      // [16][1728]
                     float* __restrict__ agg_b) {    // [16][64]
    __shared__ float att[BATCH * 12];                // K padded 10 -> 12
    const int lane = threadIdx.x;

    if (lane < BATCH) {
        const int b = lane;
        const float p0 = pooled[b * 3 + 0];
        const float p1 = pooled[b * 3 + 1];
        const float p2 = pooled[b * 3 + 2];
        float h[KATT];
        #pragma unroll
        for (int j = 0; j < KATT; ++j) {
            float v = p0 * w1[j * 3 + 0] + p1 * w1[j * 3 + 1] + p2 * w1[j * 3 + 2];
            h[j] = v > 0.0f ? v : 0.0f;
        }
        float lg[KATT];
        float mx = -1e30f;
        #pragma unroll
        for (int j = 0; j < KATT; ++j) {
            float v = b2[j];
            #pragma unroll
            for (int i = 0; i < KATT; ++i) v += h[i] * w2[j * KATT + i];
            lg[j] = v;
            mx = fmaxf(mx, v);
        }
        float e[KATT];
        float s = 0.0f;
        #pragma unroll
        for (int j = 0; j < KATT; ++j) {
            e[j] = __expf((lg[j] - mx) * TEMP_INV);
            s += e[j];
        }
        const float inv = 1.0f / s;
        #pragma unroll
        for (int j = 0; j < KATT; ++j) att[b * 12 + j] = e[j] * inv;
        att[b * 12 + 10] = 0.0f;
        att[b * 12 + 11] = 0.0f;
    }
    __syncthreads();

    // A fragment: 16(batch) x 4 fp32.  lane m = M, half selects K pair.
    const int m    = lane & 15;
    const int half = lane >> 4;
    const int kb   = half * 2;
    v2f a[3];
    #pragma unroll
    for (int g = 0; g < 3; ++g) {
        a[g].x = att[m * 12 + 4 * g + kb + 0];
        a[g].y = att[m * 12 + 4 * g + kb + 1];
    }

    int tile = blockIdx.x;
    const float* src;
    float* dst;
    int ncols;
    if (tile < 108) { src = wc; dst = agg_w; ncols = OUTC * KK27; }
    else            { src = bk; dst = agg_b; ncols = OUTC; tile -= 108; }
    const int n0 = tile * 16;

    v8f c = {};
    #pragma unroll
    for (int g = 0; g < 3; ++g) {
        const int k0 = 4 * g + kb;
        // Branch-free K padding: clamp the row index (address stays in
        // bounds), load unconditionally, select 0 for the padded lanes.
        const int k0c = k0     < KATT ? k0     : KATT - 1;
        const int k1c = k0 + 1 < KATT ? k0 + 1 : KATT - 1;
        const float v0 = src[k0c * ncols + n0 + m];
        const float v1 = src[k1c * ncols + n0 + m];
        v2f bf;
        bf.x = (k0     < KATT) ? v0 : 0.0f;
        bf.y = (k0 + 1 < KATT) ? v1 : 0.0f;
        c = __builtin_amdgcn_wmma_f32_16x16x4_f32(false, a[g], false, bf,
                                                  (short)0, c, false, false);
    }
    #pragma unroll
    for (int r = 0; r < 8; ++r) {
        const int b = r + half * 8;                  // batch row of C
        dst[b * ncols + n0 + m] = c[r];
    }
}

// ---------------------------------------------------------------------------
// Kernel 3: per-sample 3x3 conv as implicit GEMM on fp32 WMMA.
// grid = B*H blocks (one output row each), 224 threads = 7 waves.
// Wave w owns pixel tiles {2w, 2w+1} (16 px each) and all 4 oc-tiles.
// ---------------------------------------------------------------------------
// Branch-free im2col gather: clamp (kk, yy, xx) into bounds, load
// unconditionally, then select 0 for padded/out-of-image taps.
__device__ __forceinline__ float load_x(const float* __restrict__ xb, // x + b*3*H*W
                                        int kk, int y, int px) {
    const bool vk  = kk < KK27;
    const int  kkc = vk ? kk : KK27 - 1;
    const int  ic  = kkc / 9;
    const int  rem = kkc - ic * 9;
    const int  kh  = rem / 3;
    const int  kw  = rem - kh * 3;
    const int  yy  = y + kh - 1;
    const int  xx  = px + kw - 1;
    const bool vy  = (unsigned)yy < (unsigned)HW_H;
    const bool vx  = (unsigned)xx < (unsigned)HW_W;
    const int  yyc = vy ? yy : 0;
    const int  xxc = vx ? xx : 0;
    const float v  = xb[((size_t)ic * HW_H + yyc) * HW_W + xxc];
    return (vk && vy && vx) ? v : 0.0f;
}

__global__ __launch_bounds__(224)
void dynconv_kernel(const float* __restrict__ x,
                    const float* __restrict__ agg_w,   // [16][64][27]
                    const float* __restrict__ agg_b,   // [16][64]
                    float* __restrict__ out) {         // [16][64][224][224]
    const int blk  = blockIdx.x;
    const int b    = blk / HW_H;
    const int y    = blk - b * HW_H;
    const int lane = threadIdx.x & 31;
    const int wv   = threadIdx.x >> 5;                 // 0..6
    const int m    = lane & 15;
    const int half = lane >> 4;
    const int kb   = half * 2;

    const float* xb = x + (size_t)b * CIN * HW_H * HW_W;

    // A fragments: agg_w rows for 4 oc tiles, K padded 27 -> 28 (7 groups of 4)
    v2f a[4][7];
    const float* awb = agg_w + (size_t)b * OUTC * KK27;
    #pragma unroll
    for (int ot = 0; ot < 4; ++ot) {
        const float* row = awb + (ot * 16 + m) * KK27;
        #pragma unroll
        for (int g = 0; g < 7; ++g) {
            const int k0  = 4 * g + kb;
            const int k0c = k0     < KK27 ? k0     : KK27 - 1;
            const int k1c = k0 + 1 < KK27 ? k0 + 1 : KK27 - 1;
            const float v0 = row[k0c];
            const float v1 = row[k1c];
            a[ot][g].x = (k0     < KK27) ? v0 : 0.0f;
            a[ot][g].y = (k0 + 1 < KK27) ? v1 : 0.0f;
        }
    }
    const float* bb = agg_b + b * OUTC;

    #pragma unroll
    for (int t = 0; t < 2; ++t) {
        const int px0 = (wv * 2 + t) * 16;

        // Seed accumulators with the per-oc bias (same for every column).
        v8f c[4];
        #pragma unroll
        for (int ot = 0; ot < 4; ++ot) {
            #pragma unroll
            for (int r = 0; r < 8; ++r) c[ot][r] = bb[ot * 16 + half * 8 + r];
        }

        // im2col B fragment per K-group, shared across the 4 oc tiles.
        #pragma unroll
        for (int g = 0; g < 7; ++g) {
            const int k0 = 4 * g + kb;
            v2f bf;
            bf.x = load_x(xb, k0,     y, px0 + m);
            bf.y = load_x(xb, k0 + 1, y, px0 + m);
            #pragma unroll
            for (int ot = 0; ot < 4; ++ot) {
                c[ot] = __builtin_amdgcn_wmma_f32_16x16x4_f32(
                    false, a[ot][g], false, bf, (short)0, c[ot], false, false);
            }
        }

        // Store: row M -> oc, column N -> contiguous pixels (coalesced).
        // Non-temporal: output is write-once 205 MB; keep x/agg_w in cache.
        #pragma unroll
        for (int ot = 0; ot < 4; ++ot) {
            #pragma unroll
            for (int r = 0; r < 8; ++r) {
                const int oc = ot * 16 + half * 8 + r;
                float* dst = out + (((size_t)b * OUTC + oc) * HW_H + y) * HW_W + px0 + m;
                __builtin_nontemporal_store(c[ot][r], dst);
            }
        }
    }
}

// ---------------------------------------------------------------------------
extern "C" void kernel_launch(void* const* d_in, const int* in_sizes, int n_in,
                              void* d_out, int out_size, void* d_ws, size_t ws_size,
                              hipStream_t stream) {
    const float* x  = (const float*)d_in[0];   // [16][3][224][224]
    const float* w1 = (const float*)d_in[1];   // [10][3]
    const float* w2 = (const float*)d_in[2];   // [10][10]
    const float* b2 = (const float*)d_in[3];   // [10]
    const float* wc = (const float*)d_in[4];   // [10][64][3][3][3] = [10][1728]
    const float* bk = (const float*)d_in[5];   // [10][64]
    float* out = (float*)d_out;                // [16][64][224][224]

    float* ws     = (float*)d_ws;
    float* pooled = ws;                        // 48 floats
    float* agg_w  = ws + 64;                   // 16*1728 = 27648 floats
    float* agg_b  = ws + 64 + BATCH * OUTC * KK27;  // 16*64 floats

    pool_kernel<<<BATCH * CIN, 256, 0, stream>>>(x, pooled);
    attn_agg_kernel<<<112, 32, 0, stream>>>(pooled, w1, w2, b2, wc, bk, agg_w, agg_b);
    dynconv_kernel<<<BATCH * HW_H, 224, 0, stream>>>(x, agg_w, agg_b, out);
}